// InvConv1x1Conditional_7885559955516
// MI455X (gfx1250) — compile-verified
//
#include <hip/hip_runtime.h>

typedef __attribute__((ext_vector_type(2))) float v2f;
typedef __attribute__((ext_vector_type(8))) float v8f;

#define BN   32
#define CN   128
#define HWN  4096
#define CC   (CN * CN)          // 16384
#define OUTN (BN * CN * HWN)    // 16777216

// ---------------- Kernel 1: global average pool of `condition` -> pooled[B][C]
// Row staged HBM -> LDS via gfx1250 async-to-LDS DMA, then reduced from LDS.
__global__ void pool_kernel(const float* __restrict__ cond, float* __restrict__ pooled) {
    __shared__ float srow[HWN];        // 16 KB: one (b,c) row
    __shared__ float sred[256];
    const int row = blockIdx.x;        // b*C + c
    const unsigned long long src = (unsigned long long)(cond + (size_t)row * HWN);
    const unsigned int lds0 = (unsigned int)(uintptr_t)&srow[0];
    #pragma unroll
    for (int i = 0; i < 4; ++i) {
        unsigned int voff = (threadIdx.x + i * 256u) * 16u;   // byte offset in row
        unsigned int ldst = lds0 + voff;                      // per-lane LDS dest
        asm volatile("global_load_async_to_lds_b128 %0, %1, %2"
                     :: "v"(ldst), "v"(voff), "s"(src) : "memory");
    }
    asm volatile("s_wait_asynccnt 0" ::: "memory");
    __syncthreads();
    float s = 0.0f;
    const float4* sf4 = (const float4*)srow;
    #pragma unroll
    for (int i = 0; i < 4; ++i) {
        float4 v = sf4[threadIdx.x + i * 256];
        s += v.x + v.y + v.z + v.w;
    }
    sred[threadIdx.x] = s;
    __syncthreads();
    for (int st = 128; st > 0; st >>= 1) {
        if (threadIdx.x < st) sred[threadIdx.x] += sred[threadIdx.x + st];
        __syncthreads();
    }
    if (threadIdx.x == 0) pooled[row] = sred[0] * (1.0f / (float)HWN);
}

// ---------------- Kernel 2: wmat[b,o,i] = pooled[b] . w_lin[o*C+i] + b_lin + I
__global__ void condnet_kernel(const float* __restrict__ pooled,
                               const float* __restrict__ w_lin,
                               const float* __restrict__ b_lin,
                               float* __restrict__ wmat,
                               float* __restrict__ out_scalar) {
    __shared__ float ps[BN][CN + 1];   // padded: column reads conflict-free
    __shared__ float wr[8][CN];
    const int tid = threadIdx.x;
    for (int i = tid; i < BN * CN; i += 256) ps[i >> 7][i & 127] = pooled[i];
    const int j0 = blockIdx.x * 8;
    for (int i = tid; i < 8 * CN; i += 256)
        wr[i >> 7][i & 127] = w_lin[(size_t)(j0 + (i >> 7)) * CN + (i & 127)];
    __syncthreads();
    const int b  = tid & 31;
    const int jj = tid >> 5;
    const int j  = j0 + jj;
    float acc = 0.0f;
    #pragma unroll 4
    for (int k = 0; k < CN; ++k) acc += ps[b][k] * wr[jj][k];
    const int o = j >> 7, i = j & 127;
    acc += b_lin[j] + ((o == i) ? 1.0f : 0.0f);
    wmat[(size_t)b * CC + j] = acc;
    if (blockIdx.x == 0 && tid == 0) *out_scalar = 0.0f;  // zero log_det accumulator
}

// ---------------- Kernel 3: per-sample LU (partial pivoting) -> sum log|pivot|
__global__ void slogdet_kernel(const float* __restrict__ wmat,
                               float* __restrict__ out_scalar) {
    extern __shared__ float A[];       // 128 * 129 floats (dynamic: > 64KB static cap)
    __shared__ float red[128];
    __shared__ int  ridx[128];
    const int t = threadIdx.x;
    const int b = blockIdx.x;
    const float* src = wmat + (size_t)b * CC;
    for (int i = t; i < CC; i += 128) A[(i >> 7) * 129 + (i & 127)] = src[i];
    __syncthreads();
    float logsum = 0.0f;
    for (int k = 0; k < CN; ++k) {
        red[t]  = (t >= k) ? fabsf(A[t * 129 + k]) : -1.0f;
        ridx[t] = t;
        __syncthreads();
        for (int s = 64; s > 0; s >>= 1) {
            if (t < s && red[t + s] > red[t]) { red[t] = red[t + s]; ridx[t] = ridx[t + s]; }
            __syncthreads();
        }
        const int kp = ridx[0];
        if (kp != k) {                  // threads = columns, swap rows k <-> kp
            float tmp = A[k * 129 + t];
            A[k * 129 + t]  = A[kp * 129 + t];
            A[kp * 129 + t] = tmp;
        }
        __syncthreads();
        const float pivot = A[k * 129 + k];
        if (t == 0) logsum += logf(fabsf(pivot));
        if (t > k) {
            const float factor = A[t * 129 + k] / pivot;
            for (int c = k + 1; c < CN; ++c) A[t * 129 + c] -= factor * A[k * 129 + c];
        }
        __syncthreads();
    }
    // log_det = H*W * mean_b = (4096/32) * sum_b = 128 * sum_b
    if (t == 0) atomicAdd(out_scalar, logsum * 128.0f);
}

// ---------------- Kernel 4: out[b] = wmat[b] (128x128) @ inp[b] (128x4096), f32 WMMA
// LDS B-tile in k-pair-interleaved layout: Bs[kp][n][2], kp-stride 160 floats.
// -> each WMMA B fragment {B[k][n], B[k+1][n]} is ONE aligned ds_load_b64,
//    lanes 0-15 and 16-31 land on disjoint even-bank groups (conflict-free).
#define KPS 160                        // 64 kpairs * 160 floats = 40960 B LDS
__global__ void gemm_kernel(const float* __restrict__ wmat,
                            const float* __restrict__ inp,
                            float* __restrict__ out) {
    __shared__ float Bs[64 * KPS];
    const int b       = blockIdx.x >> 6;
    const int colBase = (blockIdx.x & 63) * 64;
    const int tid     = threadIdx.x;

    // cooperative stage of inp tile [K=128][N=64] into pair-interleaved LDS
    {
        const int kp0 = tid >> 4;           // 0..15
        const int c4  = (tid & 15) * 4;     // column group
        #pragma unroll
        for (int it = 0; it < 4; ++it) {
            const int kp = kp0 + it * 16;   // 0..63
            const float* g0 = inp + ((size_t)(b * CN + 2 * kp)) * HWN + colBase + c4;
            float4 r0 = *(const float4*)g0;
            float4 r1 = *(const float4*)(g0 + HWN);
            float* dst = Bs + kp * KPS + 2 * c4;        // 32B aligned
            float4 lo = {r0.x, r1.x, r0.y, r1.y};
            float4 hi = {r0.z, r1.z, r0.w, r1.w};
            *(float4*)(dst)     = lo;
            *(float4*)(dst + 4) = hi;
        }
    }
    __syncthreads();

    const int lane  = tid & 31;
    const int w     = tid >> 5;                 // wave 0..7 -> M stripe
    const int mRow  = w * 16 + (lane & 15);     // A: lane holds row M
    const int kpOff = (lane >> 4);              // upper half-lanes: next k-pair
    const int khalf = kpOff << 1;               // 0 or 2
    const float* Ap = wmat + (size_t)b * CC + (size_t)mRow * CN;
    const float* Bbase = Bs + (size_t)kpOff * KPS + 2 * (lane & 15);

    v8f acc0 = {}, acc1 = {}, acc2 = {}, acc3 = {};
    for (int k0 = 0; k0 < CN; k0 += 4) {
        v2f a = *(const v2f*)(Ap + k0 + khalf);           // global b64 (L2-resident)
        const float* Bp = Bbase + (size_t)(k0 >> 1) * KPS;
        v2f b0 = *(const v2f*)(Bp);                       // ds_load_b64, no shuffles
        v2f b1 = *(const v2f*)(Bp + 32);
        v2f b2 = *(const v2f*)(Bp + 64);
        v2f b3 = *(const v2f*)(Bp + 96);
        acc0 = __builtin_amdgcn_wmma_f32_16x16x4_f32(false, a, false, b0, (short)0, acc0, false, false);
        acc1 = __builtin_amdgcn_wmma_f32_16x16x4_f32(false, a, false, b1, (short)0, acc1, false, false);
        acc2 = __builtin_amdgcn_wmma_f32_16x16x4_f32(false, a, false, b2, (short)0, acc2, false, false);
        acc3 = __builtin_amdgcn_wmma_f32_16x16x4_f32(false, a, false, b3, (short)0, acc3, false, false);
    }

    // D layout: lanes 0-15 -> N=lane, M=r; lanes 16-31 -> N=lane-16, M=r+8
    const int n     = lane & 15;
    const int mBase = w * 16 + ((lane >> 4) << 3);
    float* outp = out + ((size_t)b * CN + mBase) * HWN + colBase + n;
    #pragma unroll
    for (int r = 0; r < 8; ++r) {
        outp[(size_t)r * HWN +  0] = acc0[r];
        outp[(size_t)r * HWN + 16] = acc1[r];
        outp[(size_t)r * HWN + 32] = acc2[r];
        outp[(size_t)r * HWN + 48] = acc3[r];
    }
}

extern "C" void kernel_launch(void* const* d_in, const int* in_sizes, int n_in,
                              void* d_out, int out_size, void* d_ws, size_t ws_size,
                              hipStream_t stream) {
    (void)in_sizes; (void)n_in; (void)out_size; (void)ws_size;
    const float* inp   = (const float*)d_in[0];
    const float* cond  = (const float*)d_in[1];
    const float* w_lin = (const float*)d_in[2];
    const float* b_lin = (const float*)d_in[3];
    float* out = (float*)d_out;
    float* ws  = (float*)d_ws;
    float* pooled = ws;            // 4096 floats
    float* wmat   = ws + 4096;     // 32*128*128 floats (2 MB)

    pool_kernel   <<<BN * CN, 256, 0, stream>>>(cond, pooled);
    condnet_kernel<<<CC / 8,  256, 0, stream>>>(pooled, w_lin, b_lin, wmat, out + OUTN);
    slogdet_kernel<<<BN, 128, 128 * 129 * sizeof(float), stream>>>(wmat, out + OUTN);
    gemm_kernel   <<<BN * 64, 256, 0, stream>>>(wmat, inp, out);
}